// GNNBranch_65687229825042
// MI455X (gfx1250) — compile-verified
//
#include <hip/hip_runtime.h>

#define N_NODES 50000
#define N_EDGES 800000
#define D 128
#define BN_EPS 1e-5f
#define ROW_TILES_PER_WAVE 5                    // 80 rows per wave; 50000 = 80*625

typedef __attribute__((ext_vector_type(2))) float v2f;
typedef __attribute__((ext_vector_type(8))) float v8f;

// ---------------------------------------------------------------- utilities
__global__ void fill_zero_kernel(float* __restrict__ p, int n) {
  int i = blockIdx.x * blockDim.x + threadIdx.x;
  if (i < n) p[i] = 0.0f;
}

// deg[dst[e]] += 1 for every real edge (self-loop contribution folded later)
__global__ void deg_kernel(const int* __restrict__ dst, float* __restrict__ deg) {
  int e = blockIdx.x * blockDim.x + threadIdx.x;
  if (e < N_EDGES) atomicAdd(&deg[dst[e]], 1.0f);
}

// in-place: deg -> dinv = rsqrt(deg + 1)   (deg+1 >= 1, so always valid)
__global__ void dinv_kernel(float* __restrict__ deg) {
  int n = blockIdx.x * blockDim.x + threadIdx.x;
  if (n < N_NODES) deg[n] = rsqrtf(deg[n] + 1.0f);
}

// ------------------------------------------------- WMMA GEMM: H = A @ W
// A: [N_NODES, D] row-major, W: [D, D] row-major, H: [N_NODES, D]
// Block = 256 threads = 8 waves; wave w owns cols 16w..16w+15.
// Each wave computes 5 row-tiles (80 rows): per k-step it loads ONE B
// fragment of W and reuses it across 5 WMMAs (5x less W traffic, denser
// wmma mix). K-loop: 32 steps x 5 = 160 x V_WMMA_F32_16X16X4_F32 per wave.
//
// ISA f32 fragment layouts (CDNA5 ISA §7.12.2):
//  A 16x4:  lanes 0-15 -> M=lane, VGPR0=K0/VGPR1=K1; lanes 16-31 -> K2/K3
//  B 4x16:  row striped across lanes: VGPR0 = {K0 | K2}, VGPR1 = {K1 | K3}
//  C/D 16x16: VGPR r -> M=r (lanes 0-15) / M=r+8 (lanes 16-31), N=lane%16
__global__ __launch_bounds__(256) void gemm_wmma_f32(
    const float* __restrict__ A, const float* __restrict__ W,
    float* __restrict__ H) {
  const int lane = threadIdx.x & 31;
  const int wave = threadIdx.x >> 5;  // 0..7 -> column tile
  const int hl   = lane >> 4;         // wave half: shifts K by 2 (A) / row pair (B)
  const int lr   = lane & 15;         // M (for A) or N (for B/C/D)
  const int row0 = blockIdx.x * (16 * ROW_TILES_PER_WAVE);
  const int col0 = wave << 4;

  v8f acc[ROW_TILES_PER_WAVE];
#pragma unroll
  for (int t = 0; t < ROW_TILES_PER_WAVE; ++t) acc[t] = (v8f){};

  const float* arow = A + (size_t)(row0 + lr) * D + 2 * hl;
  const float* wcol = W + (size_t)(2 * hl) * D + col0 + lr;

#pragma unroll
  for (int k = 0; k < D; k += 4) {
    v2f b;
    b.x = wcol[(size_t)k * D];             // W row K = k+2*hl
    b.y = wcol[(size_t)(k + 1) * D];       // W row K = k+2*hl+1
#pragma unroll
    for (int t = 0; t < ROW_TILES_PER_WAVE; ++t) {
      v2f a = *(const v2f*)(arow + (size_t)(16 * t) * D + k);
      acc[t] = __builtin_amdgcn_wmma_f32_16x16x4_f32(
          /*neg_a=*/false, a, /*neg_b=*/false, b,
          /*c_mod=*/(short)0, acc[t], /*reuse_a=*/false, /*reuse_b=*/false);
    }
  }

  float* hout = H + col0 + lr;
#pragma unroll
  for (int t = 0; t < ROW_TILES_PER_WAVE; ++t)
#pragma unroll
    for (int r = 0; r < 8; ++r)
      hout[(size_t)(row0 + 16 * t + r + 8 * hl) * D] = acc[t][r];
}

// ------------------------------------------- message passing (scatter-add)
// One wave32 per (edge or self-loop). Each lane gathers a contiguous float4
// (32 lanes x 16B = full 128-float row, perfectly coalesced, L2-resident)
// and scatter-adds with 4 f32 atomics.
__global__ __launch_bounds__(256) void scatter_kernel(
    const int* __restrict__ src, const int* __restrict__ dst,
    const float* __restrict__ dinv, const float* __restrict__ h,
    float* __restrict__ agg) {
  const int lane = threadIdx.x & 31;
  const int wid = blockIdx.x * (blockDim.x >> 5) + (threadIdx.x >> 5);
  if (wid >= N_EDGES + N_NODES) return;

  int s, d;
  float w;
  if (wid < N_EDGES) {
    s = src[wid];
    d = dst[wid];
    w = dinv[s] * dinv[d];
  } else {                       // appended self-loop: src = dst = node
    s = d = wid - N_EDGES;
    float di = dinv[s];
    w = di * di;
  }

  float4 v = ((const float4*)(h + (size_t)s * D))[lane];
  float* o = agg + (size_t)d * D + lane * 4;
  atomicAdd(o + 0, v.x * w);
  atomicAdd(o + 1, v.y * w);
  atomicAdd(o + 2, v.z * w);
  atomicAdd(o + 3, v.w * w);
}

// ------------------------------------------------- fused bias + BN(eval) + ReLU
__global__ void bn_relu_kernel(const float* __restrict__ agg,
                               const float* __restrict__ bias,
                               const float* __restrict__ g,
                               const float* __restrict__ be,
                               const float* __restrict__ rm,
                               const float* __restrict__ rv,
                               float* __restrict__ out) {
  int i = blockIdx.x * blockDim.x + threadIdx.x;
  if (i >= N_NODES * D) return;
  int c = i & (D - 1);
  float inv = g[c] * rsqrtf(rv[c] + BN_EPS);
  float v = (agg[i] + bias[c]) * inv + (be[c] - rm[c] * inv);
  out[i] = v > 0.0f ? v : 0.0f;
}

// --------------------------------------------------------------- dispatcher
extern "C" void kernel_launch(void* const* d_in, const int* in_sizes, int n_in,
                              void* d_out, int out_size, void* d_ws, size_t ws_size,
                              hipStream_t stream) {
  const float* x  = (const float*)d_in[0];
  const int* ei   = (const int*)d_in[1];       // [2, E] flat: src row then dst row
  const float* W1 = (const float*)d_in[2];
  const float* b1 = (const float*)d_in[3];
  const float* g1 = (const float*)d_in[4];
  const float* be1= (const float*)d_in[5];
  const float* rm1= (const float*)d_in[6];
  const float* rv1= (const float*)d_in[7];
  const float* W2 = (const float*)d_in[8];
  const float* b2 = (const float*)d_in[9];
  const float* g2 = (const float*)d_in[10];
  const float* be2= (const float*)d_in[11];
  const float* rm2= (const float*)d_in[12];
  const float* rv2= (const float*)d_in[13];

  const int* src = ei;
  const int* dst = ei + N_EDGES;

  // workspace layout (floats): dinv[N] | bufH[N*D] | bufA[N*D]   (~51.4 MB)
  float* dinv = (float*)d_ws;
  float* bufH = dinv + N_NODES;
  float* bufA = bufH + (size_t)N_NODES * D;
  float* out  = (float*)d_out;

  const int NT = 256;
  const int ND = N_NODES * D;                       // 6.4M
  const int scatterWaves = N_EDGES + N_NODES;       // 850000
  const int scatterBlocks = (scatterWaves + 7) / 8; // 8 waves/block
  const int gemmBlocks = N_NODES / (16 * ROW_TILES_PER_WAVE); // 625, exact

  // --- normalization: deg -> dinv
  fill_zero_kernel<<<(N_NODES + NT - 1) / NT, NT, 0, stream>>>(dinv, N_NODES);
  deg_kernel<<<(N_EDGES + NT - 1) / NT, NT, 0, stream>>>(dst, dinv);
  dinv_kernel<<<(N_NODES + NT - 1) / NT, NT, 0, stream>>>(dinv);

  // --- layer 1
  gemm_wmma_f32<<<gemmBlocks, NT, 0, stream>>>(x, W1, bufH);
  fill_zero_kernel<<<(ND + NT - 1) / NT, NT, 0, stream>>>(bufA, ND);
  scatter_kernel<<<scatterBlocks, NT, 0, stream>>>(src, dst, dinv, bufH, bufA);
  bn_relu_kernel<<<(ND + NT - 1) / NT, NT, 0, stream>>>(bufA, b1, g1, be1, rm1, rv1, bufA);

  // --- layer 2 (stream order lets us reuse the two big buffers)
  gemm_wmma_f32<<<gemmBlocks, NT, 0, stream>>>(bufA, W2, bufH);
  fill_zero_kernel<<<(ND + NT - 1) / NT, NT, 0, stream>>>(bufA, ND);
  scatter_kernel<<<scatterBlocks, NT, 0, stream>>>(src, dst, dinv, bufH, bufA);
  bn_relu_kernel<<<(ND + NT - 1) / NT, NT, 0, stream>>>(bufA, b2, g2, be2, rm2, rv2, out);
}